// BEVHead_46557445489045
// MI455X (gfx1250) — compile-verified
//
#include <hip/hip_runtime.h>
#include <hip/hip_bf16.h>
#include <stdint.h>

// ---------------------------------------------------------------------------
// BEV keypoint head for MI455X (gfx1250).
// Pipeline: guider*score -> NMS (5 fused 7x7 maxpools via async LDS tiles,
//           b128 interior + b32 halo) -> deterministic compaction ->
//           per-batch 64KB-LDS bitonic top-k -> gathers
// ---------------------------------------------------------------------------

#define HH 512
#define WW 512
#define HWSZ (HH * WW)
#define TILE 32
#define HALO 3
#define IN_T (TILE + 2 * HALO)   // 38 halo rows
#define LDST 44                  // padded LDS row stride: 176B = 11*16B (quad-aligned rows)
#define CAP 8192                 // candidate cap per batch (64KB LDS sort)

// ---- CDNA5 async global->LDS copies (per-lane) ----------------------------
__device__ __forceinline__ void async_load_b32(unsigned lds_off, const float* gaddr) {
    asm volatile("global_load_async_to_lds_b32 %0, %1, off"
                 :: "v"(lds_off), "v"((unsigned long long)(uintptr_t)gaddr)
                 : "memory");
}
__device__ __forceinline__ void async_load_b128(unsigned lds_off, const float* gaddr) {
    // 16B-aligned global and LDS addresses required.
    asm volatile("global_load_async_to_lds_b128 %0, %1, off"
                 :: "v"(lds_off), "v"((unsigned long long)(uintptr_t)gaddr)
                 : "memory");
}
__device__ __forceinline__ void wait_async0() {
#if __has_builtin(__builtin_amdgcn_s_wait_asynccnt)
    __builtin_amdgcn_s_wait_asynccnt(0);
#else
    asm volatile("s_wait_asynccnt 0x0" ::: "memory");
#endif
}
__device__ __forceinline__ unsigned lds_off_of(const void* p) {
    // flat LDS pointer low 32 bits == wave-relative LDS byte offset
    return (unsigned)(uintptr_t)p;
}

// ---- K0: score = raw * (bev[:,2] > 0) -> d_out score2d --------------------
__global__ void score_guider_kernel(const float* __restrict__ bev,
                                    const float* __restrict__ raw,
                                    float* __restrict__ score, int total) {
    int i = blockIdx.x * blockDim.x + threadIdx.x;
    if (i >= total) return;
    int b = i / HWSZ;
    int p = i - b * HWSZ;
    float g = (bev[(size_t)b * 7 * HWSZ + 2 * HWSZ + p] > 0.f) ? 1.f : 0.f;
    score[i] = raw[i] * g;
}

// ---- fused 7x7 maxpool over a 32x32 tile, async LDS staging ---------------
// Tile layout: local col lc = gx - x0 + 4, lc in [1,38]; interior lc=4..35 is
// 16B-aligned in LDS (row stride 176B) and in global (x0 % 32 == 0).
// VAR 0: in=score            -> out0 = M    = (score == pool) ? 1 : 0
// VAR 1: in=M,   score       -> out0 = SUPP = (pool > 0);  out1 = S = supp?0:score
// VAR 2: in=S,   suppIn,M    -> out0 = M'   = M | ((S==pool) & !supp)
template <int VAR>
__global__ void __launch_bounds__(256)
nms_maxpool_kernel(const float* __restrict__ in,
                   const float* __restrict__ score,
                   const float* __restrict__ suppIn,
                   float* __restrict__ out0,
                   float* __restrict__ out1) {
    __shared__ float t_in[IN_T * LDST];
    __shared__ float t_rm[IN_T * LDST];
    const int b  = blockIdx.z;
    const int x0 = blockIdx.x * TILE;
    const int y0 = blockIdx.y * TILE;
    const float* inb = in + (size_t)b * HWSZ;
    const int tid = threadIdx.x;

    // Stage halo tile: per row 8 aligned b128 quads (interior) + 6 b32 (halo).
    // Clamp-to-edge == -inf padding for max-pooling (clamped cells duplicate
    // values already inside the window).
    const int NTASK = IN_T * 14;   // 532
    for (int i = tid; i < NTASK; i += 256) {
        int ty = i / 14, r = i - ty * 14;
        int gy = y0 - HALO + ty; gy = gy < 0 ? 0 : (gy > HH - 1 ? HH - 1 : gy);
        const float* rowp = inb + (size_t)gy * WW;
        if (r < 8) {
            int gx = x0 + r * 4;                       // always in [0, WW)
            async_load_b128(lds_off_of(&t_in[ty * LDST + 4 + r * 4]), rowp + gx);
        } else {
            int j = r - 8;                             // 0..5
            int lc = (j < 3) ? (1 + j) : (36 + (j - 3));
            int gx = x0 - 4 + lc;                      // lc=1 -> x0-3, lc=36 -> x0+32
            gx = gx < 0 ? 0 : (gx > WW - 1 ? WW - 1 : gx);
            async_load_b32(lds_off_of(&t_in[ty * LDST + lc]), rowp + gx);
        }
    }
    wait_async0();
    __syncthreads();

    // Row pass: rm[ty][ox] = max of t_in[ty][ox+1 .. ox+7].
    for (int i = tid; i < IN_T * TILE; i += 256) {
        int ty = i / TILE, ox = i - ty * TILE;
        const float* r = &t_in[ty * LDST + ox + 1];
        float m = r[0];
#pragma unroll
        for (int d = 1; d < 7; ++d) m = fmaxf(m, r[d]);
        t_rm[ty * LDST + ox] = m;
    }
    __syncthreads();

    // Column pass + epilogue.
    for (int i = tid; i < TILE * TILE; i += 256) {
        int oy = i / TILE, ox = i - oy * TILE;
        float m = t_rm[oy * LDST + ox];
#pragma unroll
        for (int d = 1; d < 7; ++d) m = fmaxf(m, t_rm[(oy + d) * LDST + ox]);
        size_t p = (size_t)b * HWSZ + (size_t)(y0 + oy) * WW + (x0 + ox);
        float center = t_in[(oy + HALO) * LDST + (ox + 4)];
        if (VAR == 0) {
            out0[p] = (center == m) ? 1.f : 0.f;
        } else if (VAR == 1) {
            float sup = (m > 0.f) ? 1.f : 0.f;
            out0[p] = sup;
            out1[p] = (sup > 0.f) ? 0.f : score[p];
        } else {
            float nm  = (center == m) ? 1.f : 0.f;
            float sup = suppIn[p];
            float old = out0[p];
            out0[p] = ((old != 0.f) || (nm != 0.f && sup == 0.f)) ? 1.f : 0.f;
        }
    }
}

// ---- counting / prefix / ordered compaction -------------------------------
__global__ void count_kernel(const float* __restrict__ M, const float* __restrict__ score,
                             int* __restrict__ tileCnt) {
    int b = blockIdx.y, t = blockIdx.x;
    int base = b * HWSZ + t * 1024 + threadIdx.x * 4;
    __shared__ int cnt;
    if (threadIdx.x == 0) cnt = 0;
    __syncthreads();
    int c = 0;
#pragma unroll
    for (int j = 0; j < 4; ++j) {
        int p = base + j;
        if (M[p] != 0.f && score[p] > 0.f) c++;
    }
    atomicAdd(&cnt, c);
    __syncthreads();
    if (threadIdx.x == 0) tileCnt[b * gridDim.x + t] = cnt;
}

__global__ void scan_kernel(const int* __restrict__ tileCnt, int* __restrict__ tileOff,
                            int* __restrict__ totalC, int tiles, int B) {
    int b = threadIdx.x;
    if (b < B) {
        int run = 0;
        for (int t = 0; t < tiles; ++t) {
            tileOff[b * tiles + t] = run;
            run += tileCnt[b * tiles + t];
        }
        totalC[b] = run < CAP ? run : CAP;
    }
}

__global__ void compact_kernel(const float* __restrict__ M, const float* __restrict__ score,
                               const int* __restrict__ tileOff,
                               float* __restrict__ vals, int* __restrict__ idxs) {
    int b = blockIdx.y, t = blockIdx.x;
    const float* Mb = M + (size_t)b * HWSZ;
    const float* Sb = score + (size_t)b * HWSZ;
    int base = t * 1024 + threadIdx.x * 4;
    int hits[4];
    int c = 0;
#pragma unroll
    for (int j = 0; j < 4; ++j) {
        int p = base + j;
        if (Mb[p] != 0.f && Sb[p] > 0.f) hits[c++] = p;
    }
    __shared__ int pre[256];
    pre[threadIdx.x] = c;
    __syncthreads();
    for (int off = 1; off < 256; off <<= 1) {
        int v = pre[threadIdx.x];
        int add = (threadIdx.x >= (unsigned)off) ? pre[threadIdx.x - off] : 0;
        __syncthreads();
        pre[threadIdx.x] = v + add;
        __syncthreads();
    }
    int start = tileOff[b * gridDim.x + t] + (pre[threadIdx.x] - c);
    for (int j = 0; j < c; ++j) {
        int pos = start + j;
        if (pos < CAP) {
            vals[(size_t)b * CAP + pos] = Sb[hits[j]];
            idxs[(size_t)b * CAP + pos] = hits[j];
        }
    }
}

// ---- per-batch LDS bitonic top-k (descending value, ascending index ties) --
__global__ void __launch_bounds__(1024)
sort_select_kernel(const float* __restrict__ vals, const int* __restrict__ idxs,
                   const int* __restrict__ totalC, const float* __restrict__ M,
                   const float* __restrict__ score, int* __restrict__ selIdx, int K) {
    __shared__ unsigned long long key[CAP];   // 64 KB
    int b = blockIdx.x;
    int C = totalC[b];
    for (int i = threadIdx.x; i < CAP; i += 1024) {
        unsigned long long k = 0ull;            // sentinel sorts last
        if (i < C) {
            unsigned u = __float_as_uint(vals[(size_t)b * CAP + i]);
            u ^= (u & 0x80000000u) ? 0xFFFFFFFFu : 0x80000000u;  // monotone map
            unsigned lo = ~(unsigned)idxs[(size_t)b * CAP + i];  // smaller idx wins ties
            k = ((unsigned long long)u << 32) | lo;
        }
        key[i] = k;
    }
    __syncthreads();
    for (int kk = 2; kk <= CAP; kk <<= 1) {
        for (int j = kk >> 1; j > 0; j >>= 1) {
            for (int i = threadIdx.x; i < CAP; i += 1024) {
                int ixj = i ^ j;
                if (ixj > i) {
                    unsigned long long a = key[i], c = key[ixj];
                    bool desc = ((i & kk) == 0);
                    if (desc ? (a < c) : (a > c)) { key[i] = c; key[ixj] = a; }
                }
            }
            __syncthreads();
        }
    }
    int lim = C < K ? C : K;
    for (int s = threadIdx.x; s < lim; s += 1024)
        selIdx[b * K + s] = (int)(~(unsigned)(key[s] & 0xFFFFFFFFull));
    __syncthreads();
    if (threadIdx.x == 0 && C < K) {  // lax.top_k fill: -1 entries at lowest free idx
        const float* Mb = M + (size_t)b * HWSZ;
        const float* Sb = score + (size_t)b * HWSZ;
        int w = C, p = 0;
        while (w < K && p < HWSZ) {
            if (!(Mb[p] != 0.f && Sb[p] > 0.f)) selIdx[b * K + w++] = p;
            p++;
        }
        while (w < K) selIdx[b * K + w++] = 0;
    }
}

// ---- gathers: kpts, pixels, features --------------------------------------
__global__ void gather_kernel(const float* __restrict__ bev, const float* __restrict__ feat,
                              const int* __restrict__ selIdx,
                              float* __restrict__ kpts, float* __restrict__ feas,
                              float* __restrict__ pix, int CF, int K) {
    int s = blockIdx.x, b = blockIdx.y, c = threadIdx.x;
    int idx = selIdx[b * K + s];
    feas[((size_t)b * CF + c) * K + s] = feat[((size_t)b * CF + c) * HWSZ + idx];
    if (c == 0) {
        int u = idx / WW, v = idx - u * WW;
        const float* bb = bev + (size_t)b * 7 * HWSZ;
        float* kp = kpts + ((size_t)b * K + s) * 4;
        kp[0] = bb[3 * HWSZ + idx];
        kp[1] = bb[4 * HWSZ + idx];
        kp[2] = 0.f;
        kp[3] = 1.f;
        float* pp = pix + ((size_t)b * K + s) * 2;
        pp[0] = (float)u;
        pp[1] = (float)v;
    }
}

// ---------------------------------------------------------------------------
extern "C" void kernel_launch(void* const* d_in, const int* in_sizes, int n_in,
                              void* d_out, int out_size, void* d_ws, size_t ws_size,
                              hipStream_t stream) {
    const float* bev  = (const float*)d_in[0];
    const float* raw  = (const float*)d_in[1];
    const float* feat = (const float*)d_in[2];

    const int B  = in_sizes[1] / HWSZ;                       // 4
    const int CF = in_sizes[2] / (B * HWSZ);                 // 128
    const int K  = (out_size / B - HWSZ) / (4 + CF + 2);     // 512 (derived, capture-safe)
    const int tiles = HWSZ / 1024;                           // 256

    // output layout (flat concat, float)
    float* score2d = (float*)d_out;
    float* kpts    = score2d + (size_t)B * HWSZ;
    float* feas    = kpts + (size_t)B * K * 4;
    float* pix     = feas + (size_t)B * CF * K;

    // workspace layout
    char* w = (char*)d_ws;
    float* M    = (float*)w;               w += (size_t)B * HWSZ * 4;
    float* SUPP = (float*)w;               w += (size_t)B * HWSZ * 4;
    float* S    = (float*)w;               w += (size_t)B * HWSZ * 4;
    int* tileCnt = (int*)w;                w += (size_t)B * tiles * 4;
    int* tileOff = (int*)w;                w += (size_t)B * tiles * 4;
    int* totalC  = (int*)w;                w += (size_t)B * 4;
    w = (char*)(((uintptr_t)w + 15) & ~(uintptr_t)15);
    float* vals  = (float*)w;              w += (size_t)B * CAP * 4;
    int* idxs    = (int*)w;                w += (size_t)B * CAP * 4;
    int* selIdx  = (int*)w;

    // 1) guided score (== score2d output)
    {
        int total = B * HWSZ;
        score_guider_kernel<<<(total + 255) / 256, 256, 0, stream>>>(bev, raw, score2d, total);
    }

    dim3 pg(WW / TILE, HH / TILE, B);
    // 2) initial mask M
    nms_maxpool_kernel<0><<<pg, 256, 0, stream>>>(score2d, nullptr, nullptr, M, nullptr);
    // 3) two NMS iterations
    for (int it = 0; it < 2; ++it) {
        nms_maxpool_kernel<1><<<pg, 256, 0, stream>>>(M, score2d, nullptr, SUPP, S);
        nms_maxpool_kernel<2><<<pg, 256, 0, stream>>>(S, nullptr, SUPP, M, nullptr);
    }

    // 4) deterministic compaction of (M & score>0)
    count_kernel<<<dim3(tiles, B), 256, 0, stream>>>(M, score2d, tileCnt);
    scan_kernel<<<1, 64, 0, stream>>>(tileCnt, tileOff, totalC, tiles, B);
    compact_kernel<<<dim3(tiles, B), 256, 0, stream>>>(M, score2d, tileOff, vals, idxs);

    // 5) per-batch LDS bitonic sort + top-K select (+ lax.top_k fill semantics)
    sort_select_kernel<<<B, 1024, 0, stream>>>(vals, idxs, totalC, M, score2d, selIdx, K);

    // 6) gathers
    gather_kernel<<<dim3(K, B), CF, 0, stream>>>(bev, feat, selIdx, kpts, feas, pix, CF, K);
}